// JointNet_34196529611142
// MI455X (gfx1250) — compile-verified
//
#include <hip/hip_runtime.h>

// ---------------------------------------------------------------------------
// RNN-T joint network, restructured:
//   out[b,t,u,v] = A_enc[b*T+t, v] + A_dec[b*U+u, v]
//   A_enc = gelu(enc) @ W[:, :512]^T + bias     (1200 x 640, K=512)
//   A_dec = gelu(dec) @ W[:, 512:]^T            ( 404 x 640, K=512)
// Pipeline:
//   1) gelu_fill: precompute gelu(enc), gelu(dec) once (padded rows -> 0),
//      so the WMMA K-loop is pure loads + v_wmma (no libm tanh in the loop).
//   2) gemm_wmma_f32: V_WMMA_F32_16X16X4_F32, one wave per 16x16 tile.
//      Output buffers are row-padded to multiples of 16 so the D-store
//      epilogue is unconditional (no per-row EXEC juggling).
//   3) bcast_add_nt: streams the 310 MB output with non-temporal b128 stores
//      (write-once data must not thrash the 192 MB L2); A-tables stay in L2.
// ---------------------------------------------------------------------------

typedef __attribute__((ext_vector_type(2))) float v2f;
typedef __attribute__((ext_vector_type(4))) float v4f;
typedef __attribute__((ext_vector_type(8))) float v8f;

#define B_    4
#define T_    300
#define U_    101
#define DENC  512
#define DIN   1024
#define V_    640

#define M_ENC (B_ * T_)              // 1200 (multiple of 16)
#define M_DEC (B_ * U_)              // 404
#define M_DEC_PAD 416                // padded to multiple of 16

__device__ __forceinline__ float gelu_tanh(float x) {
    // tanh-approximate GELU (matches jax.nn.gelu(approximate=True))
    const float kAlpha = 0.7978845608028654f;   // sqrt(2/pi)
    const float kBeta  = 0.044715f;
    float inner = kAlpha * (x + kBeta * x * x * x);
    return 0.5f * x * (1.0f + tanhf(inner));
}

// G[i] = gelu(X[i]) for i < n, 0 for n <= i < nPad (zero-fills pad rows).
__global__ void __launch_bounds__(256)
gelu_fill(const float* __restrict__ X, float* __restrict__ G, int n, int nPad)
{
    int i = blockIdx.x * blockDim.x + threadIdx.x;
    if (i < nPad) G[i] = (i < n) ? gelu_tanh(X[i]) : 0.0f;
}

// Out[m, n] = sum_{k<512} G[m, k] * W[n, kOffset + k]   (+ bias[n])
//   G:   Mpad x 512 row-major (gelu pre-applied, pad rows zero)
//   W:   640 x 1024 row-major
//   Out: Mpad x 640 row-major (pad rows written but never read)
// One wave computes one 16x16 tile; threadIdx.y picks the N-tile.
// Inner loop: two b64 fragment loads + one V_WMMA_F32_16X16X4_F32.
__global__ void __launch_bounds__(128)
gemm_wmma_f32(const float* __restrict__ G,
              const float* __restrict__ W,
              const float* __restrict__ bias,
              float* __restrict__ Out,
              int kOffset)
{
    const int lane = threadIdx.x;          // 0..31
    const int half = lane >> 4;            // lanes 16-31 hold K+2 (A and B frags)
    const int l16  = lane & 15;
    const int n0   = (blockIdx.x * blockDim.y + threadIdx.y) * 16;
    const int m0   = blockIdx.y * 16;

    // A 16x4 frag: lane = M row (l16), VGPR j = K = k0 + 2*half + j
    const v2f* __restrict__ ga =
        (const v2f*)(G + (size_t)(m0 + l16) * DENC + 2 * half);
    // B 4x16 frag: lane = N col (l16), VGPR j = K = k0 + 2*half + j;
    // B[k][n] = W[n][kOffset + k], contiguous along k.
    const v2f* __restrict__ wb =
        (const v2f*)(W + (size_t)(n0 + l16) * DIN + kOffset + 2 * half);

    // C/D layout: VGPR r = row (m0 + 8*half + r), col (n0 + l16).
    // Bias depends only on the column -> broadcast into all 8 accumulators.
    v8f acc;
    {
        float c0 = bias ? bias[n0 + l16] : 0.0f;
        #pragma unroll
        for (int r = 0; r < 8; ++r) acc[r] = c0;
    }

    #pragma unroll 8
    for (int k0 = 0; k0 < DENC; k0 += 4) {
        v2f a = ga[k0 >> 1];
        v2f b = wb[k0 >> 1];
        // 8-arg f32 pattern: (neg_a, A, neg_b, B, c_mod, C, reuse_a, reuse_b)
        acc = __builtin_amdgcn_wmma_f32_16x16x4_f32(
            false, a, false, b, (short)0, acc, false, false);
    }

    // Unconditional, coalesced b32 stores (16 lane-columns x 4B per row).
    float* __restrict__ orow = Out + (size_t)(m0 + 8 * half) * V_ + n0 + l16;
    #pragma unroll
    for (int r = 0; r < 8; ++r) orow[(size_t)r * V_] = acc[r];
}

// out[row, :] = A_enc[row/U, :] + A_dec[(row/(T*U))*U + row%U, :]
// One block per (b,t,u) row; 160 threads x float4 == 640 floats exactly.
// A tables (~4 MB) are L2-resident; output is write-once -> non-temporal.
__global__ void __launch_bounds__(160)
bcast_add_nt(const float* __restrict__ Ae,
             const float* __restrict__ Ad,
             float* __restrict__ Out)
{
    const int row = blockIdx.x;            // 0 .. B*T*U-1
    const int u   = row % U_;
    const int bt  = row / U_;
    const int b   = bt / T_;

    const v4f* __restrict__ e = (const v4f*)(Ae + (size_t)bt * V_);
    const v4f* __restrict__ d = (const v4f*)(Ad + ((size_t)b * U_ + u) * V_);
    v4f* __restrict__ o       = (v4f*)(Out + (size_t)row * V_);

    const int i = threadIdx.x;             // 0..159
    v4f r = e[i] + d[i];
    __builtin_nontemporal_store(r, o + i);
}

extern "C" void kernel_launch(void* const* d_in, const int* in_sizes, int n_in,
                              void* d_out, int out_size, void* d_ws, size_t ws_size,
                              hipStream_t stream) {
    const float* enc = (const float*)d_in[0];   // (B, T, 512)
    const float* dec = (const float*)d_in[1];   // (B, U, 512)
    const float* Wfc = (const float*)d_in[2];   // (640, 1024)
    const float* bfc = (const float*)d_in[3];   // (640,)
    float* out = (float*)d_out;                 // (B, T, U, 640)

    // Workspace layout (floats): Ae | Ad(padded) | Ge | Gd(padded)  (~7.4 MB)
    float* Ae = (float*)d_ws;                       // M_ENC * V_
    float* Ad = Ae + (size_t)M_ENC * V_;            // M_DEC_PAD * V_
    float* Ge = Ad + (size_t)M_DEC_PAD * V_;        // M_ENC * 512
    float* Gd = Ge + (size_t)M_ENC * DENC;          // M_DEC_PAD * 512

    // 1) GELU precompute (pad rows of Gd zero-filled).
    {
        const int nEnc = M_ENC * DENC;              // 614400
        gelu_fill<<<dim3((nEnc + 255) / 256), dim3(256), 0, stream>>>(
            enc, Ge, nEnc, nEnc);
        const int nDec    = M_DEC * DENC;           // 206848
        const int nDecPad = M_DEC_PAD * DENC;       // 212992
        gelu_fill<<<dim3((nDecPad + 255) / 256), dim3(256), 0, stream>>>(
            dec, Gd, nDec, nDecPad);
    }

    // 2) WMMA GEMMs.
    const dim3 blk(32, 4);                          // 4 waves = 4 N-tiles/block
    const int  nTileBlocks = V_ / (16 * 4);         // 10

    // A_enc = Ge @ W[:, :512]^T + bias   (M = 1200)
    gemm_wmma_f32<<<dim3(nTileBlocks, M_ENC / 16), blk, 0, stream>>>(
        Ge, Wfc, bfc, Ae, 0);

    // A_dec = Gd @ W[:, 512:]^T          (rows 404..415 are pad, never read)
    gemm_wmma_f32<<<dim3(nTileBlocks, M_DEC_PAD / 16), blk, 0, stream>>>(
        Gd, Wfc, nullptr, Ad, DENC);

    // 3) Streaming broadcast-add: one block per (b,t,u) row.
    bcast_add_nt<<<dim3(B_ * T_ * U_), dim3(160), 0, stream>>>(Ae, Ad, out);
}